// UnProcessData_85469849191059
// MI455X (gfx1250) — compile-verified
//
#include <hip/hip_runtime.h>

typedef float v2f __attribute__((ext_vector_type(2)));
typedef float v8f __attribute__((ext_vector_type(8)));
typedef unsigned int u32;
typedef u32 v4u __attribute__((ext_vector_type(4)));
typedef int  v8i __attribute__((ext_vector_type(8)));
typedef int  v4i __attribute__((ext_vector_type(4)));

#define N_SEQ 256
#define T_LEN 1024
#define TP1   1025
#define J_NUM 24
#define NTJ   (N_SEQ * TP1 * J_NUM)          // 6,297,600
#define ROT_WS_FLOATS (N_SEQ * TP1 * 2)      // cos/sin half-angle per (n,t)

// ---------------------------------------------------------------------------
// Inclusive prefix-sum of 1024 f32 values in LDS, using V_WMMA_F32_16X16X4_F32.
// Data mapped to a 16x64 matrix M[row][col] = xs[col*16 + row]; column-wise
// prefix = L(16x16 lower-tri ones) @ M, done as 4 chained K=4 WMMAs per
// 16-column tile (waves 0..3, one tile each), then a 64-wide serial carry
// scan and a carry-apply pass. Block must be 256 threads.
// ---------------------------------------------------------------------------
__device__ __forceinline__ void block_scan1024_wmma(float* xs, float* tot, int tid) {
    const int wave = tid >> 5;
    const int lane = tid & 31;
    __syncthreads();
    if (wave < 4) {
        const int w   = wave;        // tile index: columns 16w .. 16w+15
        const int hi  = lane >> 4;   // 0: lanes 0-15, 1: lanes 16-31
        const int col = lane & 15;   // B/D column (N index)
        const int row = lane & 15;   // A row (M index)
        v8f c = {};
#pragma unroll
        for (int k = 0; k < 4; ++k) {
            const int kb = 4 * k + 2 * hi;   // K index for VGPR0 of this half
            v2f a, b;
            // A = L chunk (16x4): L[row][K] = (K <= row)
            a.x = (kb     <= row) ? 1.0f : 0.0f;
            a.y = (kb + 1 <= row) ? 1.0f : 0.0f;
            // B = M chunk (4x16): M[K][col] = xs[(16w+col)*16 + K]
            b.x = xs[(16 * w + col) * 16 + kb];
            b.y = xs[(16 * w + col) * 16 + kb + 1];
            c = __builtin_amdgcn_wmma_f32_16x16x4_f32(false, a, false, b,
                                                      (short)0, c, false, false);
        }
        // D layout: VGPR v -> P[v + 8*hi][col]; write partial prefixes in place
#pragma unroll
        for (int v = 0; v < 8; ++v) {
            xs[(16 * w + col) * 16 + v + 8 * hi] = c[v];
        }
        if (hi) tot[16 * w + col] = c[7];    // column total = P[15][col]
    }
    __syncthreads();
    if (tid == 0) {                           // exclusive scan of 64 column totals
        float carry = 0.0f;
        for (int cc = 0; cc < 64; ++cc) {
            float tmp = tot[cc];
            tot[cc]   = carry;
            carry    += tmp;
        }
    }
    __syncthreads();
    for (int i = tid; i < 1024; i += 256) xs[i] += tot[i >> 4];
    __syncthreads();
}

// ---------------------------------------------------------------------------
// Kernel A: per-sequence scans + yaw-rotation table + root trajectory.
// grid = N_SEQ blocks of 256 threads.
// ---------------------------------------------------------------------------
__global__ void __launch_bounds__(256)
seq_scan_kernel(const float* __restrict__ root_rvel,   // (N,T,1,1)
                const float* __restrict__ root_vel,    // (N,T,1,2)
                float* __restrict__ rot2,              // ws: (N,TP1,2) {cos,sin}
                float* __restrict__ pos3)              // ws: (N,TP1,3)
{
    const int n   = blockIdx.x;
    const int tid = threadIdx.x;
    __shared__ float xs[1024];
    __shared__ float tot[64];
    __shared__ float rc[TP1];
    __shared__ float rs[TP1];
    __shared__ float vr[3][1024];

    // ---- yaw angle cumsum ----
    for (int t = tid; t < T_LEN; t += 256) xs[t] = root_rvel[n * T_LEN + t];
    block_scan1024_wmma(xs, tot, tid);   // xs[t] = r[t+1] (inclusive cumsum)

    // ---- axis-angle (0, r, 0) -> quat (w, 0, y, 0), small-angle branch ----
    for (int tt = tid; tt < TP1; tt += 256) {
        float r    = (tt == 0) ? 0.0f : xs[tt - 1];
        float ang  = fabsf(r);
        float half = 0.5f * ang;
        float k    = (ang < 1e-6f) ? (0.5f - ang * ang * (1.0f / 48.0f))
                                   : (sinf(half) / ang);
        float qw = cosf(half);
        float qy = r * k;
        rc[tt] = qw;
        rs[tt] = qy;
        rot2[(n * TP1 + tt) * 2 + 0] = qw;
        rot2[(n * TP1 + tt) * 2 + 1] = qy;
    }
    __syncthreads();

    // ---- v3w = R_y(r[t]) * (vx, 0, vz) via quat sandwich (zeros folded) ----
    for (int t = tid; t < T_LEN; t += 256) {
        float vx = root_vel[(n * T_LEN + t) * 2 + 0];
        float vz = root_vel[(n * T_LEN + t) * 2 + 1];
        float w = rc[t], y = rs[t];
        float ox = w * vx + y * vz;
        float oz = w * vz - y * vx;
        vr[0][t] =  ox * w + oz * y;
        vr[1][t] =  0.0f;                 // exact (vy == 0 in the reference path)
        vr[2][t] = -ox * y + oz * w;
    }
    __syncthreads();

    // ---- trajectory cumsum per component ----
    for (int comp = 0; comp < 3; ++comp) {
        for (int t = tid; t < T_LEN; t += 256) xs[t] = vr[comp][t];
        block_scan1024_wmma(xs, tot, tid);
        for (int t = tid; t < T_LEN; t += 256)
            pos3[(n * TP1 + t + 1) * 3 + comp] = xs[t];
        if (tid == 0) pos3[(n * TP1) * 3 + comp] = 0.0f;
        __syncthreads();
    }
}

// ---------------------------------------------------------------------------
// TDM: async copy of `nfloats` contiguous f32 from global to LDS (1-D tile).
// D# built per CDNA5 ISA; 6-arg builtin form
// (amdgpu-toolchain clang-23: g0 v4u, g1 v8i, g2 v4i, g3 v4i, extra v8i, cpol).
// ---------------------------------------------------------------------------
__device__ __forceinline__ void tdm_copy_to_lds(const void* gptr, u32 lds_off, u32 nfloats) {
    unsigned long long ga = (unsigned long long)gptr;
    v4u g0;
    g0.x = 1u;                                              // count=1 valid descriptor
    g0.y = lds_off;                                         // lds_addr (bytes)
    g0.z = (u32)(ga & 0xFFFFFFFFu);                         // global_addr[31:0]
    g0.w = (u32)((ga >> 32) & 0x01FFFFFFu) | (2u << 30);    // global_addr[56:32] | type=2
    v8i g1;
    g1[0] = (int)(2u << 16);                                // data_size = 4 bytes
    g1[1] = (int)((nfloats & 0xFFFFu) << 16);               // tensor_dim0[15:0] @ bits 63:48
    g1[2] = (int)((nfloats >> 16) | (1u << 16));            // tensor_dim0[31:16], tensor_dim1=1
    g1[3] = (int)(nfloats << 16);                           // tile_dim0 @ bits 127:112
    g1[4] = 0;                                              // tile_dim1=0, tile_dim2=0 (unused)
    g1[5] = (int)nfloats;                                   // tensor_dim0_stride[31:0]
    g1[6] = 0;
    g1[7] = 0;
    v4i z4 = {0, 0, 0, 0};
    v8i z8 = {0, 0, 0, 0, 0, 0, 0, 0};
    __builtin_amdgcn_tensor_load_to_lds(g0, g1, z4, z4, z8, 0);
}

// ---------------------------------------------------------------------------
// Kernel B: bulk per-joint work (bandwidth-bound). One thread per (n,t,j).
// Block of 256 stages its contiguous local_rot/local_pos slices into LDS via
// the Tensor Data Mover, then computes 6D->quat, quat compose, rotate, add.
// ---------------------------------------------------------------------------
__global__ void __launch_bounds__(256)
joint_kernel(const float* __restrict__ local_pos,   // (N,TP1,J,3)
             const float* __restrict__ local_rot,   // (N,TP1,J,6)
             const float* __restrict__ rot2,        // ws
             const float* __restrict__ pos3,        // ws
             float* __restrict__ outPos,            // (N,TP1,J,3)
             float* __restrict__ outRot)            // (N,TP1,J,4)
{
    const int tid  = threadIdx.x;
    const int base = blockIdx.x * 256;

    __shared__ float sRot[256 * 6];
    __shared__ float sPos[256 * 3];

    if (tid < 32) {   // wave 0 issues both async DMAs, waits on TENSORcnt
        tdm_copy_to_lds(local_rot + (size_t)base * 6,
                        (u32)(unsigned long long)&sRot[0], 256u * 6u);
        tdm_copy_to_lds(local_pos + (size_t)base * 3,
                        (u32)(unsigned long long)&sPos[0], 256u * 3u);
        __builtin_amdgcn_s_wait_tensorcnt(0);
    }
    __syncthreads();

    const int idx = base + tid;               // < NTJ, fits in int
    const int nt  = idx / J_NUM;              // n*TP1 + t

    // yaw quaternion (w, 0, y, 0) and trajectory, broadcast over J (L2-resident)
    const float rcw = rot2[nt * 2 + 0];
    const float rsy = rot2[nt * 2 + 1];
    const float txp = pos3[nt * 3 + 0];
    const float typ = pos3[nt * 3 + 1];
    const float tzp = pos3[nt * 3 + 2];

    // ---- 6D -> rotation matrix (Gram-Schmidt) ----
    const float a1x = sRot[tid * 6 + 0], a1y = sRot[tid * 6 + 1], a1z = sRot[tid * 6 + 2];
    const float a2x = sRot[tid * 6 + 3], a2y = sRot[tid * 6 + 4], a2z = sRot[tid * 6 + 5];
    float inv1 = 1.0f / sqrtf(a1x * a1x + a1y * a1y + a1z * a1z);
    float b1x = a1x * inv1, b1y = a1y * inv1, b1z = a1z * inv1;
    float d   = b1x * a2x + b1y * a2y + b1z * a2z;
    float px2 = a2x - d * b1x, py2 = a2y - d * b1y, pz2 = a2z - d * b1z;
    float inv2 = 1.0f / sqrtf(px2 * px2 + py2 * py2 + pz2 * pz2);
    float b2x = px2 * inv2, b2y = py2 * inv2, b2z = pz2 * inv2;
    float b3x = b1y * b2z - b1z * b2y;
    float b3y = b1z * b2x - b1x * b2z;
    float b3z = b1x * b2y - b1y * b2x;
    // M rows: b1, b2, b3
    const float m00 = b1x, m01 = b1y, m02 = b1z;
    const float m10 = b2x, m11 = b2y, m12 = b2z;
    const float m20 = b3x, m21 = b3y, m22 = b3z;

    // ---- matrix -> quaternion (argmax candidate) ----
    float qa0 = sqrtf(fmaxf(1.0f + m00 + m11 + m22, 0.0f));
    float qa1 = sqrtf(fmaxf(1.0f + m00 - m11 - m22, 0.0f));
    float qa2 = sqrtf(fmaxf(1.0f - m00 + m11 - m22, 0.0f));
    float qa3 = sqrtf(fmaxf(1.0f - m00 - m11 + m22, 0.0f));
    int   bi  = 0; float bq = qa0;
    if (qa1 > bq) { bq = qa1; bi = 1; }
    if (qa2 > bq) { bq = qa2; bi = 2; }
    if (qa3 > bq) { bq = qa3; bi = 3; }
    float lw, lx, ly, lz;
    if (bi == 0)      { lw = qa0 * qa0; lx = m21 - m12; ly = m02 - m20; lz = m10 - m01; }
    else if (bi == 1) { lw = m21 - m12; lx = qa1 * qa1; ly = m10 + m01; lz = m02 + m20; }
    else if (bi == 2) { lw = m02 - m20; lx = m10 + m01; ly = qa2 * qa2; lz = m21 + m12; }
    else              { lw = m10 - m01; lx = m20 + m02; ly = m21 + m12; lz = qa3 * qa3; }
    float s = 1.0f / (2.0f * fmaxf(bq, 0.1f));
    lw *= s; lx *= s; ly *= s; lz *= s;

    // ---- global_rot = standardize( (rcw,0,rsy,0) * (lw,lx,ly,lz) ) ----
    float qw = rcw * lw - rsy * ly;
    float qx = rcw * lx + rsy * lz;
    float qy = rcw * ly + rsy * lw;
    float qz = rcw * lz - rsy * lx;
    if (qw < 0.0f) { qw = -qw; qx = -qx; qy = -qy; qz = -qz; }

    // ---- global_pos = quat_apply(yaw, local_pos) + trajectory ----
    const float ppx = sPos[tid * 3 + 0];
    const float ppy = sPos[tid * 3 + 1];
    const float ppz = sPos[tid * 3 + 2];
    float ow = -rsy * ppy;
    float ox =  rcw * ppx + rsy * ppz;
    float oy =  rcw * ppy;
    float oz =  rcw * ppz - rsy * ppx;
    float gx =  ox * rcw + oz * rsy + txp;
    float gy = -ow * rsy + oy * rcw + typ;
    float gz = -ox * rsy + oz * rcw + tzp;

    // ---- streamed outputs (non-temporal: written once, never re-read) ----
    __builtin_nontemporal_store(gx, outPos + (size_t)idx * 3 + 0);
    __builtin_nontemporal_store(gy, outPos + (size_t)idx * 3 + 1);
    __builtin_nontemporal_store(gz, outPos + (size_t)idx * 3 + 2);
    __builtin_nontemporal_store(qw, outRot + (size_t)idx * 4 + 0);
    __builtin_nontemporal_store(qx, outRot + (size_t)idx * 4 + 1);
    __builtin_nontemporal_store(qy, outRot + (size_t)idx * 4 + 2);
    __builtin_nontemporal_store(qz, outRot + (size_t)idx * 4 + 3);
}

extern "C" void kernel_launch(void* const* d_in, const int* in_sizes, int n_in,
                              void* d_out, int out_size, void* d_ws, size_t ws_size,
                              hipStream_t stream) {
    (void)in_sizes; (void)n_in; (void)out_size; (void)ws_size;
    const float* root_rvel = (const float*)d_in[0];   // (N,T,1,1)
    const float* local_pos = (const float*)d_in[1];   // (N,T+1,J,3)
    const float* local_rot = (const float*)d_in[2];   // (N,T+1,J,6)
    const float* root_vel  = (const float*)d_in[3];   // (N,T,1,2)

    float* rot2 = (float*)d_ws;                        // N*TP1*2 floats
    float* pos3 = rot2 + ROT_WS_FLOATS;                // N*TP1*3 floats

    float* outPos = (float*)d_out;                     // N*TP1*J*3 floats
    float* outRot = outPos + (size_t)NTJ * 3;          // N*TP1*J*4 floats

    seq_scan_kernel<<<N_SEQ, 256, 0, stream>>>(root_rvel, root_vel, rot2, pos3);
    joint_kernel<<<NTJ / 256, 256, 0, stream>>>(local_pos, local_rot, rot2, pos3,
                                                outPos, outRot);
}